// PrototypeMemory_41497974014037
// MI455X (gfx1250) — compile-verified
//
#include <hip/hip_runtime.h>
#include <hip/hip_bf16.h>
#include <math.h>

#define DD 2048
#define KK 16384
#define THRESH 0.5f
#define DECAY 0.99f

typedef __attribute__((ext_vector_type(2))) float v2f;
typedef __attribute__((ext_vector_type(8))) float v8f;

// ---------------- K1: inv norm of z ----------------
__global__ void k1_znorm(const float* __restrict__ z, float* __restrict__ ws) {
    __shared__ float red[256];
    float s = 0.f;
    for (int i = threadIdx.x; i < DD; i += 256) { float v = z[i]; s += v * v; }
    red[threadIdx.x] = s;
    __syncthreads();
    for (int off = 128; off > 0; off >>= 1) {
        if (threadIdx.x < off) red[threadIdx.x] += red[threadIdx.x + off];
        __syncthreads();
    }
    if (threadIdx.x == 0) ws[0] = rsqrtf(red[0]);
}

// ---------------- K2: logits via V_WMMA_F32_16X16X4_F32 ----------------
// 8 waves / block = 2 tiles x 4 D-segments. Each wave computes a partial
// 16-row x 512-col accumulation: A = 16x4 f32 chunk of the prototype tile
// (ISA layout: lanes 0-15 rows, K=0..1 half0 / K=2..3 half1), B = z chunk
// replicated across all 16 columns, so C[m,n] = partial dot(p_m, z) for
// every n; column 0 lives in lanes 0 and 16. Partials (dots + sumsq) are
// combined across the 4 D-segment waves through LDS.
__global__ void __launch_bounds__(256) k2_logits(const float* __restrict__ z,
                                                 const float* __restrict__ protos,
                                                 const float* __restrict__ tempp,
                                                 float* __restrict__ ws) {
    __shared__ float zl[DD];
    __shared__ float pdots[2][4][16];   // [tileInBlock][dseg][row]
    __shared__ float psq[2][4][16];
    for (int i = threadIdx.x; i < DD; i += 256) zl[i] = z[i];
    __syncthreads();

    const int wave = threadIdx.x >> 5;
    const int lane = threadIdx.x & 31;
    const int half = lane >> 4;        // 0: K=0,1  1: K=2,3
    const int r    = lane & 15;        // row within tile
    const int tib  = wave >> 2;        // tile in block: 0..1
    const int dseg = wave & 3;         // D segment: 0..3 (512 cols each)
    const int tile = blockIdx.x * 2 + tib;
    const int row  = tile * 16 + r;
    const int cbase = dseg * (DD / 4); // 512 cols = 128 chunks of 4

    const float* arow = protos + (size_t)row * DD + dseg * 512 + 2 * half;
    const float* zseg = zl + dseg * 512;

    v8f acc = {};
    float sq = 0.f;
    for (int c = 0; c < 128; ++c) {
        v2f a = *(const v2f*)(arow + 4 * c);
        v2f b;
        b.x = zseg[4 * c + 2 * half];
        b.y = zseg[4 * c + 2 * half + 1];
        acc = __builtin_amdgcn_wmma_f32_16x16x4_f32(
            /*neg_a=*/false, a, /*neg_b=*/false, b,
            /*c_mod=*/(short)0, acc, /*reuse_a=*/false, /*reuse_b=*/false);
        sq += a.x * a.x + a.y * a.y;
    }
    (void)cbase;
    // combine the two K-halves of the partial sum-of-squares (rows = lane&15)
    sq += __shfl_xor(sq, 16, 32);

    // column N=0 of C: lane 0 holds rows 0..7 (VGPR v = row v),
    // lane 16 holds rows 8..15.
    if (lane == 0)  { for (int v = 0; v < 8; ++v) pdots[tib][dseg][v]     = acc[v]; }
    if (lane == 16) { for (int v = 0; v < 8; ++v) pdots[tib][dseg][8 + v] = acc[v]; }
    if (lane < 16)  psq[tib][dseg][r] = sq;
    __syncthreads();

    // combine D-segment partials: 32 outputs per block (2 tiles x 16 rows)
    if (threadIdx.x < 32) {
        const int t2 = threadIdx.x >> 4;
        const int r2 = threadIdx.x & 15;
        float dot = 0.f, sqa = 0.f;
        for (int d = 0; d < 4; ++d) {
            dot += pdots[t2][d][r2];
            sqa += psq[t2][d][r2];
        }
        const int tout = blockIdx.x * 2 + t2;
        float invz = ws[0];
        ws[8 + tout * 16 + r2] = tempp[0] * dot * invz * rsqrtf(sqa);
    }
}

// ---------------- K3: gate (max logit, u, softmax denom, argmin usage) ----
__global__ void k3_gate(const float* __restrict__ usages,
                        const float* __restrict__ beta,
                        const float* __restrict__ gamma,
                        float* __restrict__ ws, int* __restrict__ wsi) {
    __shared__ float rmax[256];
    __shared__ float rmin[256];
    __shared__ int   rmini[256];
    const int tid = threadIdx.x;

    float mx = -1e30f;
    for (int i = tid; i < KK; i += 256) mx = fmaxf(mx, ws[8 + i]);
    rmax[tid] = mx;

    float mn = 1e30f; int mni = 0x7fffffff;
    for (int i = tid; i < KK; i += 256) {
        float v = usages[i];
        if (v < mn) { mn = v; mni = i; }   // strict '<' keeps first occurrence
    }
    rmin[tid] = mn; rmini[tid] = mni;
    __syncthreads();

    for (int off = 128; off > 0; off >>= 1) {
        if (tid < off) {
            rmax[tid] = fmaxf(rmax[tid], rmax[tid + off]);
            if (rmin[tid + off] < rmin[tid] ||
                (rmin[tid + off] == rmin[tid] && rmini[tid + off] < rmini[tid])) {
                rmin[tid] = rmin[tid + off]; rmini[tid] = rmini[tid + off];
            }
        }
        __syncthreads();
    }
    const float maxl = rmax[0];
    const int   amin = rmini[0];
    __syncthreads();

    float se = 0.f;
    for (int i = tid; i < KK; i += 256) se += expf(ws[8 + i] - maxl);
    rmax[tid] = se;
    __syncthreads();
    for (int off = 128; off > 0; off >>= 1) {
        if (tid < off) rmax[tid] += rmax[tid + off];
        __syncthreads();
    }

    if (tid == 0) {
        float u = 1.f / (1.f + expf(-((-maxl - beta[0]) / gamma[0])));
        ws[1] = maxl;
        ws[2] = rmax[0];                 // sum exp
        ws[3] = u;
        ws[4] = (u >= THRESH) ? 1.f : 0.f;
        wsi[5] = amin;
    }
}

// ---------------- K4: fused prototype update + logits2 dot products -------
// One wave per prototype row (8 waves / block).
__global__ void __launch_bounds__(256) k4_update(const float* __restrict__ z,
                                                 const float* __restrict__ protos,
                                                 const float* __restrict__ usages,
                                                 const float* __restrict__ tempp,
                                                 const float* __restrict__ ws,
                                                 const int* __restrict__ wsi,
                                                 float* __restrict__ out_protos,
                                                 float* __restrict__ out_usages,
                                                 float* __restrict__ logits2) {
    __shared__ float zl[DD];
    for (int i = threadIdx.x; i < DD; i += 256) zl[i] = z[i];
    __syncthreads();

    const int wave = threadIdx.x >> 5;
    const int lane = threadIdx.x & 31;
    const int row  = blockIdx.x * 8 + wave;

    const float invz   = ws[0];
    const float maxl   = ws[1];
    const float sumexp = ws[2];
    const float u      = ws[3];
    const bool  evict  = (ws[4] != 0.f);
    const int   idx    = wsi[5];
    const float temp   = tempp[0];

    float w1 = 0.f, w2 = 0.f, new_u;
    int  src = row;
    bool tail = false;

    if (evict) {
        if (row == KK - 1) { tail = true; new_u = 1.f; }
        else {
            src = (row < idx) ? row : row + 1;
            new_u = usages[src];
        }
    } else {
        float lg    = ws[8 + row];
        float y     = expf(lg - maxl) / sumexp;
        float delta = y * (1.f - u);
        float us    = usages[row];
        w1 = delta / (delta + us);
        w2 = us / (us + delta);
        new_u = us + delta;
    }

    const float* prow = protos + (size_t)src * DD;
    float* orow = out_protos + (size_t)row * DD;

    float dzp = 0.f, dpp = 0.f;
    for (int j = lane; j < DD; j += 32) {
        float zv = zl[j];
        float val;
        if (evict) val = tail ? zv : prow[j];
        else       val = zv * w1 + prow[j] * w2;
        orow[j] = val;
        dzp += val * zv;
        dpp += val * val;
    }
    for (int off = 16; off > 0; off >>= 1) {
        dzp += __shfl_xor(dzp, off, 32);
        dpp += __shfl_xor(dpp, off, 32);
    }
    if (lane == 0) {
        out_usages[row] = new_u * DECAY;
        logits2[row]    = temp * dzp * invz * rsqrtf(dpp);
    }
}

// ---------------- K5: argmax label + self cross-entropy loss --------------
__global__ void k5_loss(const float* __restrict__ logits2,
                        const float* __restrict__ ws,
                        float* __restrict__ out) {
    __shared__ float rv[256];
    __shared__ int   ri[256];
    const int tid = threadIdx.x;

    float mx = -1e30f; int mi = 0x7fffffff;
    for (int i = tid; i < KK; i += 256) {
        float v = logits2[i];
        if (v > mx) { mx = v; mi = i; }    // strict '>' keeps first occurrence
    }
    rv[tid] = mx; ri[tid] = mi;
    __syncthreads();
    for (int off = 128; off > 0; off >>= 1) {
        if (tid < off) {
            if (rv[tid + off] > rv[tid] ||
                (rv[tid + off] == rv[tid] && ri[tid + off] < ri[tid])) {
                rv[tid] = rv[tid + off]; ri[tid] = ri[tid + off];
            }
        }
        __syncthreads();
    }
    const float maxl = rv[0];
    const int label = ri[0];
    __syncthreads();

    float se = 0.f;
    for (int i = tid; i < KK; i += 256) se += expf(logits2[i] - maxl);
    rv[tid] = se;
    __syncthreads();
    for (int off = 128; off > 0; off >>= 1) {
        if (tid < off) rv[tid] += rv[tid + off];
        __syncthreads();
    }

    if (tid == 0) {
        // loss = -(log_softmax at argmax) = log(sum exp(l - max))
        out[0] = logf(rv[0]);
        out[1] = (float)label;
        out[2] = ws[3];   // u
    }
}

extern "C" void kernel_launch(void* const* d_in, const int* in_sizes, int n_in,
                              void* d_out, int out_size, void* d_ws, size_t ws_size,
                              hipStream_t stream) {
    const float* z      = (const float*)d_in[0];
    const float* protos = (const float*)d_in[1];
    const float* usages = (const float*)d_in[2];
    const float* beta   = (const float*)d_in[3];
    const float* gamma  = (const float*)d_in[4];
    const float* temp   = (const float*)d_in[5];

    float* out        = (float*)d_out;
    float* ws         = (float*)d_ws;          // [0..7] scalars, [8..8+K) logits
    int*   wsi        = (int*)d_ws;
    float* logits2    = ws + 8 + KK;           // [8+K .. 8+2K)
    float* out_protos = out + 3;
    float* out_usages = out + 3 + (size_t)KK * DD;

    k1_znorm <<<1,        256, 0, stream>>>(z, ws);
    k2_logits<<<KK / 32,  256, 0, stream>>>(z, protos, temp, ws);   // 2 tiles/block
    k3_gate  <<<1,        256, 0, stream>>>(usages, beta, gamma, ws, wsi);
    k4_update<<<KK / 8,   256, 0, stream>>>(z, protos, usages, temp, ws, wsi,
                                            out_protos, out_usages, logits2);
    k5_loss  <<<1,        256, 0, stream>>>(logits2, ws, out);
}